// MetricLoss_56882546868460
// MI455X (gfx1250) — compile-verified
//
#include <hip/hip_runtime.h>

typedef float v2f __attribute__((ext_vector_type(2)));
typedef float v8f __attribute__((ext_vector_type(8)));

#define K_CLUST 32
#define CCH 16
#define NIMG 8
#define IMG_PIX (512 * 512)
#define BLOCKS_PER_IMG 64
#define PIX_PER_BLOCK (IMG_PIX / BLOCKS_PER_IMG)        // 4096
#define WAVES_PER_BLOCK 8
#define PIX_PER_WAVE (PIX_PER_BLOCK / WAVES_PER_BLOCK)  // 512

// Workspace layout (float offsets)
#define WS_SUMS 0      // 8 * 512
#define WS_CNTS 4096   // 8 * 32
#define WS_PULL 4352   // 8
#define WS_PUSH 4360   // 8
#define WS_CENT 4368   // 8 * 512
#define WS_ZERO_FLOATS (WS_PUSH + 8)

// ---------------------------------------------------------------------------
// Kernel 1: per-image segment sums + counts via one-hot GEMM with
// V_WMMA_F32_16X16X4_F32 (full f32, exact vs reference segment_sum).
// A (16x4, MxK) = one-hot(label); B (4x16, KxN) = 4 pixels x 16 channels.
// Label broadcast uses v_readlane_b32 (SGPR, no LDS traffic) instead of
// __shfl (which lowered to ds_bpermute_b32 + s_wait_dscnt on the hot path).
// ---------------------------------------------------------------------------
__global__ void segsum_wmma_kernel(const float* __restrict__ emb,
                                   const int* __restrict__ lab,
                                   float* __restrict__ ws) {
  __shared__ float lds_sums[K_CLUST * CCH];
  __shared__ float lds_cnts[K_CLUST];

  const int tid  = threadIdx.x;
  const int img  = blockIdx.x / BLOCKS_PER_IMG;
  const int blk  = blockIdx.x % BLOCKS_PER_IMG;
  const int lane = tid & 31;
  const int wave = tid >> 5;

  for (int i = tid; i < K_CLUST * CCH; i += blockDim.x) lds_sums[i] = 0.0f;
  if (tid < K_CLUST) lds_cnts[tid] = 0.0f;
  __syncthreads();

  const long imgPixBase = (long)img * IMG_PIX;
  const float* __restrict__ embImg = emb + imgPixBase * CCH;
  const int* __restrict__ labImg = lab + imgPixBase;
  const int wavePix0 = blk * PIX_PER_BLOCK + wave * PIX_PER_WAVE;

  const int m = lane & 15;                              // cluster row for A
  const bool loHalf = (lane < 16);
  const int off1 = (lane & 15) | ((lane & 16) << 1);    // B: ch + pixel(0|2)*16

  v8f accLo = {0.f, 0.f, 0.f, 0.f, 0.f, 0.f, 0.f, 0.f}; // clusters 0..15
  v8f accHi = {0.f, 0.f, 0.f, 0.f, 0.f, 0.f, 0.f, 0.f}; // clusters 16..31
  float cntLo = 0.0f, cntHi = 0.0f;

  for (int t = 0; t < PIX_PER_WAVE; t += 32) {
    const int myLab = labImg[wavePix0 + t + lane];       // 32 labels / wave
#pragma unroll
    for (int j = 0; j < 8; ++j) {
      const int p0 = 4 * j;
      // Wave-uniform broadcasts: v_readlane_b32 -> SGPRs (no DS ops/waits)
      const int l0 = __builtin_amdgcn_readlane(myLab, p0 + 0);
      const int l1 = __builtin_amdgcn_readlane(myLab, p0 + 1);
      const int l2 = __builtin_amdgcn_readlane(myLab, p0 + 2);
      const int l3 = __builtin_amdgcn_readlane(myLab, p0 + 3);
      // A layout (16x4 f32): V0 = K0 (lanes 0-15) / K2 (lanes 16-31); V1 = K1/K3
      const int le  = loHalf ? l0 : l2;   // even pixel for this half-wave
      const int lo_ = loHalf ? l1 : l3;   // odd pixel for this half-wave
      v2f a, ah;
      a.x = (le == m) ? 1.0f : 0.0f;
      a.y = (lo_ == m) ? 1.0f : 0.0f;
      ah.x = (le == m + 16) ? 1.0f : 0.0f;
      ah.y = (lo_ == m + 16) ? 1.0f : 0.0f;
      cntLo += a.x + a.y;
      cntHi += ah.x + ah.y;
      // B layout (4x16 f32): V0 = K0/K2 rows, V1 = K1/K3 rows, N = lane&15
      const float* __restrict__ bp = embImg + (long)(wavePix0 + t + p0) * CCH;
      v2f b;
      b.x = bp[off1];
      b.y = bp[off1 + 16];
      accLo = __builtin_amdgcn_wmma_f32_16x16x4_f32(false, a, false, b,
                                                    (short)0, accLo, false, false);
      accHi = __builtin_amdgcn_wmma_f32_16x16x4_f32(false, ah, false, b,
                                                    (short)0, accHi, false, false);
    }
  }
  // counts: fold the two half-wave pixel streams together
  cntLo += __shfl_xor(cntLo, 16, 32);
  cntHi += __shfl_xor(cntHi, 16, 32);

  // D layout: VGPR j -> cluster j (lanes 0-15) / j+8 (lanes 16-31); N = lane&15
  const int ch = lane & 15;
  const int mbase = (lane >> 4) << 3;  // 0 or 8
#pragma unroll
  for (int j = 0; j < 8; ++j) {
    atomicAdd(&lds_sums[(mbase + j) * CCH + ch], accLo[j]);
    atomicAdd(&lds_sums[(16 + mbase + j) * CCH + ch], accHi[j]);
  }
  if (lane < 16) {
    atomicAdd(&lds_cnts[lane], cntLo);
    atomicAdd(&lds_cnts[lane + 16], cntHi);
  }
  __syncthreads();

  float* __restrict__ gsums = ws + WS_SUMS + img * (K_CLUST * CCH);
  float* __restrict__ gcnts = ws + WS_CNTS + img * K_CLUST;
  for (int i = tid; i < K_CLUST * CCH; i += blockDim.x) atomicAdd(&gsums[i], lds_sums[i]);
  if (tid < K_CLUST) atomicAdd(&gcnts[tid], lds_cnts[tid]);
}

// ---------------------------------------------------------------------------
// Kernel 2: centroids + push loss + reg (one block per image; tiny)
// ---------------------------------------------------------------------------
__global__ void centroid_push_kernel(float* __restrict__ ws) {
  __shared__ float cent[K_CLUST * CCH];
  __shared__ int validS[K_CLUST];
  __shared__ float wH[WAVES_PER_BLOCK];
  __shared__ float wR[WAVES_PER_BLOCK];

  const int img = blockIdx.x;
  const int tid = threadIdx.x;
  const float* __restrict__ gsums = ws + WS_SUMS + img * (K_CLUST * CCH);
  const float* __restrict__ gcnts = ws + WS_CNTS + img * K_CLUST;

  if (tid < K_CLUST) validS[tid] = (gcnts[tid] > 0.0f) ? 1 : 0;
  __syncthreads();
  for (int i = tid; i < K_CLUST * CCH; i += blockDim.x) {
    const int k = i >> 4;
    const float cnt = gcnts[k];
    const float c = (cnt > 0.0f) ? gsums[i] / cnt : 0.0f;
    cent[i] = c;
    ws[WS_CENT + img * (K_CLUST * CCH) + i] = c;
  }
  __syncthreads();

  // push: hinged squared L1 over valid strict-upper-triangle pairs
  float sumH = 0.0f;
  for (int p = tid; p < K_CLUST * K_CLUST; p += blockDim.x) {
    const int i = p >> 5, j = p & 31;
    if (i < j && validS[i] && validS[j]) {
      float d = 0.0f;
#pragma unroll
      for (int c = 0; c < CCH; ++c) d += fabsf(cent[i * CCH + c] - cent[j * CCH + c]);
      const float h = 0.25f - d;  // PUSH_MARGIN
      if (h > 0.0f) sumH += h * h;
    }
  }
  // reg: sum of squared valid-centroid entries
  float sumR = 0.0f;
  for (int i = tid; i < K_CLUST * CCH; i += blockDim.x) {
    if (validS[i >> 4]) { const float c = cent[i]; sumR += c * c; }
  }
#pragma unroll
  for (int o = 16; o > 0; o >>= 1) {
    sumH += __shfl_xor(sumH, o, 32);
    sumR += __shfl_xor(sumR, o, 32);
  }
  if ((tid & 31) == 0) { wH[tid >> 5] = sumH; wR[tid >> 5] = sumR; }
  __syncthreads();
  if (tid == 0) {
    float H = 0.0f, R = 0.0f, nv = 0.0f;
    for (int w = 0; w < WAVES_PER_BLOCK; ++w) { H += wH[w]; R += wR[w]; }
    for (int k = 0; k < K_CLUST; ++k) nv += (float)validS[k];
    const float ncomp = nv * (nv - 1.0f) * 0.5f;
    const float push = (nv >= 2.0f) ? H / fmaxf(ncomp, 1.0f) : 0.0f;
    const float reg = R / fmaxf(nv * (float)CCH, 1.0f);
    ws[WS_PUSH + img] = push + 1e-4f * reg;  // PUSH_WEIGHT=1, REG_WEIGHT=1e-4
  }
}

// ---------------------------------------------------------------------------
// Kernel 3: pull loss (second streaming pass; emb+labels fit in 192MB L2)
// ---------------------------------------------------------------------------
__global__ void pull_kernel(const float* __restrict__ emb,
                            const int* __restrict__ lab,
                            float* __restrict__ ws) {
  __shared__ float cent[K_CLUST * CCH];
  __shared__ float wsum[WAVES_PER_BLOCK];

  const int tid = threadIdx.x;
  const int img = blockIdx.x / BLOCKS_PER_IMG;
  const int blk = blockIdx.x % BLOCKS_PER_IMG;
  for (int i = tid; i < K_CLUST * CCH; i += blockDim.x)
    cent[i] = ws[WS_CENT + img * (K_CLUST * CCH) + i];
  __syncthreads();

  const long base = (long)img * IMG_PIX + (long)blk * PIX_PER_BLOCK;
  const float4* __restrict__ e4 = (const float4*)(emb + base * CCH);
  const int* __restrict__ lb = lab + base;

  float acc = 0.0f;
  for (int p = tid; p < PIX_PER_BLOCK; p += 256) {
    const int k = lb[p];
    const float4* __restrict__ cp = (const float4*)&cent[k * CCH];
    float d = 0.0f;
#pragma unroll
    for (int q = 0; q < 4; ++q) {
      const float4 e = e4[p * 4 + q];
      const float4 c = cp[q];
      d += fabsf(e.x - c.x) + fabsf(e.y - c.y) + fabsf(e.z - c.z) + fabsf(e.w - c.w);
    }
    acc += d * d;  // PULL_MARGIN = 0, d >= 0
  }
#pragma unroll
  for (int o = 16; o > 0; o >>= 1) acc += __shfl_xor(acc, o, 32);
  if ((tid & 31) == 0) wsum[tid >> 5] = acc;
  __syncthreads();
  if (tid == 0) {
    float s = 0.0f;
    for (int w = 0; w < WAVES_PER_BLOCK; ++w) s += wsum[w];
    atomicAdd(&ws[WS_PULL + img], s);
  }
}

// ---------------------------------------------------------------------------
// Kernel 4: combine per-image losses -> scalar
// ---------------------------------------------------------------------------
__global__ void finalize_kernel(const float* __restrict__ ws, float* __restrict__ out) {
  if (threadIdx.x == 0 && blockIdx.x == 0) {
    float s = 0.0f;
    for (int i = 0; i < NIMG; ++i)
      s += ws[WS_PUSH + i] + ws[WS_PULL + i] * (1.0f / (float)IMG_PIX);
    out[0] = s * (1.0f / (float)NIMG);
  }
}

extern "C" void kernel_launch(void* const* d_in, const int* in_sizes, int n_in,
                              void* d_out, int out_size, void* d_ws, size_t ws_size,
                              hipStream_t stream) {
  (void)in_sizes; (void)n_in; (void)out_size; (void)ws_size;
  const float* emb = (const float*)d_in[0];
  const int* lab = (const int*)d_in[1];
  float* ws = (float*)d_ws;

  hipMemsetAsync(d_ws, 0, WS_ZERO_FLOATS * sizeof(float), stream);
  segsum_wmma_kernel<<<NIMG * BLOCKS_PER_IMG, 256, 0, stream>>>(emb, lab, ws);
  centroid_push_kernel<<<NIMG, 256, 0, stream>>>(ws);
  pull_kernel<<<NIMG * BLOCKS_PER_IMG, 256, 0, stream>>>(emb, lab, ws);
  finalize_kernel<<<1, 32, 0, stream>>>(ws, (float*)d_out);
}